// GCN_15479062135311
// MI455X (gfx1250) — compile-verified
//
#include <hip/hip_runtime.h>

typedef __attribute__((ext_vector_type(2))) float v2f;
typedef __attribute__((ext_vector_type(8))) float v8f;
typedef __attribute__((ext_vector_type(4))) unsigned int v4u;
typedef __attribute__((ext_vector_type(8))) int v8i;
typedef __attribute__((ext_vector_type(4))) int v4i;

#define FEAT1 128
#define FEAT2 40

__device__ __forceinline__ void atomAddF(float* p, float v) {
    __hip_atomic_fetch_add(p, v, __ATOMIC_RELAXED, __HIP_MEMORY_SCOPE_AGENT);
}

// ---- Tensor Data Mover: DMA a rows x cols fp32 tile (row-major, stride=cols) into LDS @0 ----
// D# bitfields per CDNA5 ISA (2-D tensor, higher groups zero).
// 6-arg builtin form (clang-23 / therock-10.0): (u32x4, i32x8, i32x4, i32x4, i32x8, i32 cpol).
__device__ __forceinline__ void tdm_load_tile_lds0(const void* gptr, unsigned rows, unsigned cols) {
    unsigned long long ga = (unsigned long long)(uintptr_t)gptr;
    v4u g0;
    g0[0] = 1u;                                            // count=1, user mode, no gather
    g0[1] = 0u;                                            // lds_addr = 0 (tile at LDS base)
    g0[2] = (unsigned)(ga & 0xffffffffull);                // global_addr[31:0]
    g0[3] = (unsigned)((ga >> 32) & 0x01ffffffull)         // global_addr[56:32]
          | (2u << 30);                                    // type = 2 ("image")
    v8i g1;
    g1[0] = (int)(2u << 16);                               // data_size = 2 -> 4 bytes
    g1[1] = (int)(cols << 16);                             // tensor_dim0[15:0]
    g1[2] = (int)(rows << 16);                             // dim0[31:16]=0 | tensor_dim1[15:0]
    g1[3] = (int)(cols << 16);                             // dim1[31:16]=0 | tile_dim0
    g1[4] = (int)rows;                                     // tile_dim1 | tile_dim2=0
    g1[5] = (int)cols;                                     // tensor_dim0_stride[31:0]
    g1[6] = 0;                                             // stride0[47:32] | stride1[15:0]
    g1[7] = 0;
    v4i gz4 = {0, 0, 0, 0};
    v8i gz8 = {0, 0, 0, 0, 0, 0, 0, 0};
    __builtin_amdgcn_tensor_load_to_lds(g0, g1, gz4, gz4, gz8, 0);
}

// ---------------- degree / norm ----------------
__global__ void gcn_deg_init(float* __restrict__ deg, int n) {
    int i = blockIdx.x * blockDim.x + threadIdx.x;
    if (i < n) deg[i] = 1.0f;   // self loop
}

__global__ void gcn_deg_count(const int* __restrict__ dst, float* __restrict__ deg, int e) {
    int i = blockIdx.x * blockDim.x + threadIdx.x;
    if (i < e) atomAddF(&deg[dst[i]], 1.0f);
}

__global__ void gcn_dinv(float* __restrict__ deg, int n) {
    int i = blockIdx.x * blockDim.x + threadIdx.x;
    if (i < n) {
        float d = deg[i];
        deg[i] = (d > 0.0f) ? rsqrtf(d) : 0.0f;
    }
}

// ---------------- GEMM1: H[Nx128] = X[Nx128] @ W[128x128] (WMMA f32 16x16x4) ----------------
// block = 256 threads (8 waves). Each block: 16 rows; wave w owns cols [16w,16w+16).
__global__ void gcn_gemm1_wmma(const float* __restrict__ X, const float* __restrict__ W,
                               float* __restrict__ H) {
    __shared__ float As[16 * FEAT1];
    const int m0 = blockIdx.x * 16;
    if (threadIdx.x < 32) {   // one wave issues the TDM descriptor (TDM ignores EXEC)
        tdm_load_tile_lds0(X + (size_t)m0 * FEAT1, 16, FEAT1);
        __builtin_amdgcn_s_wait_tensorcnt(0);
    }
    __syncthreads();
    // Escape As into a may-write asm: TDM wrote LDS behind the compiler's back, so plain
    // loads below must not be folded to undef, yet still lower to fast ds_load.
    asm volatile("" :: "s"(As) : "memory");

    const int wave = threadIdx.x >> 5;
    const int lane = threadIdx.x & 31;
    const int half = lane >> 4;     // 0: lanes 0-15, 1: lanes 16-31
    const int lo   = lane & 15;     // A: row M ; B/D: col N
    const int n0   = wave * 16;

    v8f acc = {};
    for (int k0 = 0; k0 < FEAT1; k0 += 4) {
        v2f a, b;
        // A 16x4 layout: lane(0-15) K=0,1 ; lane(16-31) K=2,3
        a.x = As[lo * FEAT1 + k0 + 2 * half];
        a.y = As[lo * FEAT1 + k0 + 2 * half + 1];
        // B 4x16 layout: VGPR0 rows K=0(lanes0-15)/K=2(lanes16-31), VGPR1 rows K=1/K=3
        b.x = W[(k0 + 2 * half) * FEAT1 + n0 + lo];
        b.y = W[(k0 + 2 * half + 1) * FEAT1 + n0 + lo];
        acc = __builtin_amdgcn_wmma_f32_16x16x4_f32(false, a, false, b, (short)0, acc,
                                                    false, false);
    }
    // D 16x16: VGPR r -> row r (lanes 0-15) / row r+8 (lanes 16-31), col = lo
#pragma unroll
    for (int r = 0; r < 8; ++r)
        H[(size_t)(m0 + r + 8 * half) * FEAT1 + n0 + lo] = acc[r];
}

// ---------------- GEMM2: H2[Nx40] = Hin[Nx128] @ W[128x40] ----------------
// block = 96 threads (3 waves). Wave w owns cols [16w,16w+16), guarded at 40.
__global__ void gcn_gemm2_wmma(const float* __restrict__ Hin, const float* __restrict__ W,
                               float* __restrict__ H2) {
    __shared__ float As[16 * FEAT1];
    const int m0 = blockIdx.x * 16;
    if (threadIdx.x < 32) {
        tdm_load_tile_lds0(Hin + (size_t)m0 * FEAT1, 16, FEAT1);
        __builtin_amdgcn_s_wait_tensorcnt(0);
    }
    __syncthreads();
    asm volatile("" :: "s"(As) : "memory");

    const int wave = threadIdx.x >> 5;
    const int lane = threadIdx.x & 31;
    const int half = lane >> 4;
    const int lo   = lane & 15;
    const int n0   = wave * 16;
    const int col  = n0 + lo;
    const bool inN = (col < FEAT2);

    v8f acc = {};
    for (int k0 = 0; k0 < FEAT1; k0 += 4) {
        v2f a, b;
        a.x = As[lo * FEAT1 + k0 + 2 * half];
        a.y = As[lo * FEAT1 + k0 + 2 * half + 1];
        b.x = inN ? W[(k0 + 2 * half) * FEAT2 + col] : 0.0f;
        b.y = inN ? W[(k0 + 2 * half + 1) * FEAT2 + col] : 0.0f;
        acc = __builtin_amdgcn_wmma_f32_16x16x4_f32(false, a, false, b, (short)0, acc,
                                                    false, false);
    }
    if (inN) {
#pragma unroll
        for (int r = 0; r < 8; ++r)
            H2[(size_t)(m0 + r + 8 * half) * FEAT2 + col] = acc[r];
    }
}

// ---------------- aggregation, layer 1 (128 feats, float4 chunks: 32/node) ----------------
__global__ void gcn_agg_init128(const float* __restrict__ H, const float* __restrict__ dinv,
                                const float* __restrict__ bias, float* __restrict__ out, int n) {
    long long idx = (long long)blockIdx.x * blockDim.x + threadIdx.x;
    if (idx >= (long long)n * 32) return;
    int node = (int)(idx >> 5);
    int c    = (int)(idx & 31);
    float s  = dinv[node]; s = s * s;                 // self-loop norm
    float4 h = ((const float4*)H)[idx];
    float4 b = ((const float4*)bias)[c];
    float4 o;
    o.x = b.x + h.x * s; o.y = b.y + h.y * s;
    o.z = b.z + h.z * s; o.w = b.w + h.w * s;
    ((float4*)out)[idx] = o;
}

__global__ void gcn_agg_edge128(const float* __restrict__ H, const float* __restrict__ dinv,
                                const int* __restrict__ src, const int* __restrict__ dst,
                                float* __restrict__ out, int e) {
    long long idx = (long long)blockIdx.x * blockDim.x + threadIdx.x;
    if (idx >= (long long)e * 32) return;
    int ed = (int)(idx >> 5);
    int c  = (int)(idx & 31);
    // speculative prefetch of a future edge's gather row (global_prefetch_b8)
    if (ed + 64 < e) {
        int sn = src[ed + 64];
        __builtin_prefetch(&((const float4*)H)[(size_t)sn * 32 + c], 0, 1);
    }
    int s  = src[ed], d = dst[ed];
    float nrm = dinv[s] * dinv[d];
    float4 h = ((const float4*)H)[(size_t)s * 32 + c];
    float* p = out + (size_t)d * FEAT1 + c * 4;
    atomAddF(p + 0, h.x * nrm);
    atomAddF(p + 1, h.y * nrm);
    atomAddF(p + 2, h.z * nrm);
    atomAddF(p + 3, h.w * nrm);
}

__global__ void gcn_relu128(float* __restrict__ buf, int n) {
    long long idx = (long long)blockIdx.x * blockDim.x + threadIdx.x;
    if (idx >= (long long)n * 32) return;
    float4 v = ((float4*)buf)[idx];
    v.x = fmaxf(v.x, 0.0f); v.y = fmaxf(v.y, 0.0f);
    v.z = fmaxf(v.z, 0.0f); v.w = fmaxf(v.w, 0.0f);
    ((float4*)buf)[idx] = v;
}

// ---------------- aggregation, layer 2 (40 feats, float4 chunks: 10/node) ----------------
__global__ void gcn_agg_init40(const float* __restrict__ H, const float* __restrict__ dinv,
                               const float* __restrict__ bias, float* __restrict__ out, int n) {
    long long idx = (long long)blockIdx.x * blockDim.x + threadIdx.x;
    if (idx >= (long long)n * 10) return;
    int node = (int)(idx / 10);
    int c    = (int)(idx - (long long)node * 10);
    float s  = dinv[node]; s = s * s;
    float4 h = ((const float4*)H)[idx];
    float4 b = ((const float4*)bias)[c];
    float4 o;
    o.x = b.x + h.x * s; o.y = b.y + h.y * s;
    o.z = b.z + h.z * s; o.w = b.w + h.w * s;
    ((float4*)out)[idx] = o;
}

__global__ void gcn_agg_edge40(const float* __restrict__ H, const float* __restrict__ dinv,
                               const int* __restrict__ src, const int* __restrict__ dst,
                               float* __restrict__ out, int e) {
    long long idx = (long long)blockIdx.x * blockDim.x + threadIdx.x;
    if (idx >= (long long)e * 10) return;
    int ed = (int)(idx / 10);
    int c  = (int)(idx - (long long)ed * 10);
    if (ed + 64 < e) {
        int sn = src[ed + 64];
        __builtin_prefetch(&((const float4*)H)[(size_t)sn * 10 + c], 0, 1);
    }
    int s  = src[ed], d = dst[ed];
    float nrm = dinv[s] * dinv[d];
    float4 h = ((const float4*)H)[(size_t)s * 10 + c];
    float* p = out + (size_t)d * FEAT2 + c * 4;
    atomAddF(p + 0, h.x * nrm);
    atomAddF(p + 1, h.y * nrm);
    atomAddF(p + 2, h.z * nrm);
    atomAddF(p + 3, h.w * nrm);
}

extern "C" void kernel_launch(void* const* d_in, const int* in_sizes, int n_in,
                              void* d_out, int out_size, void* d_ws, size_t ws_size,
                              hipStream_t stream) {
    const float* x  = (const float*)d_in[0];
    const int*   ei = (const int*)d_in[1];
    const float* W1 = (const float*)d_in[2];
    const float* b1 = (const float*)d_in[3];
    const float* W2 = (const float*)d_in[4];
    const float* b2 = (const float*)d_in[5];

    const int N = in_sizes[0] / FEAT1;   // 100000
    const int E = in_sizes[1] / 2;       // 1600000
    const int* src = ei;
    const int* dst = ei + E;

    float* logits = (float*)d_out;                       // [N,40]
    float* hidden = (float*)d_out + (size_t)N * FEAT2;   // [N,128]

    float* dinv = (float*)d_ws;                               // N floats (deg -> dinv in place)
    float* h1   = (float*)((char*)d_ws + (1u << 20));         // [N,128]; reused for h2 [N,40]
    float* h2   = h1;

    const int T = 256;
    dim3 blk(T);

    // degrees + symmetric norm
    gcn_deg_init<<<dim3((N + T - 1) / T), blk, 0, stream>>>(dinv, N);
    gcn_deg_count<<<dim3((E + T - 1) / T), blk, 0, stream>>>(dst, dinv, E);
    gcn_dinv<<<dim3((N + T - 1) / T), blk, 0, stream>>>(dinv, N);

    // layer 1
    gcn_gemm1_wmma<<<dim3(N / 16), dim3(256), 0, stream>>>(x, W1, h1);
    long long t1 = (long long)N * 32;
    gcn_agg_init128<<<dim3((unsigned)((t1 + T - 1) / T)), blk, 0, stream>>>(h1, dinv, b1, hidden, N);
    long long e1 = (long long)E * 32;
    gcn_agg_edge128<<<dim3((unsigned)((e1 + T - 1) / T)), blk, 0, stream>>>(h1, dinv, src, dst, hidden, E);
    gcn_relu128<<<dim3((unsigned)((t1 + T - 1) / T)), blk, 0, stream>>>(hidden, N);

    // layer 2
    gcn_gemm2_wmma<<<dim3(N / 16), dim3(96), 0, stream>>>(hidden, W2, h2);
    long long t2 = (long long)N * 10;
    gcn_agg_init40<<<dim3((unsigned)((t2 + T - 1) / T)), blk, 0, stream>>>(h2, dinv, b2, logits, N);
    long long e2 = (long long)E * 10;
    gcn_agg_edge40<<<dim3((unsigned)((e2 + T - 1) / T)), blk, 0, stream>>>(h2, dinv, src, dst, logits, E);
}